// GreedyRouter_46909632807587
// MI455X (gfx1250) — compile-verified
//
#include <hip/hip_runtime.h>
#include <stdint.h>

#define N_EXPERTS   64
#define TOP_K       8
#define TILE_TOKENS 256
#define THREADS     256
#define ROW_DW      68   // 64 data dwords + 4 pad dwords: bank-conflict-free, rows 16B-aligned (272B)

typedef float        f32x4 __attribute__((ext_vector_type(4)));
typedef unsigned int u32x4 __attribute__((ext_vector_type(4)));
typedef int          i32x4 __attribute__((ext_vector_type(4)));
typedef int          i32x8 __attribute__((ext_vector_type(8)));

// Branchless sorted-descending insertion stage (v_max_u32 / v_min_u32 pair)
#define INS_STAGE(s, k) { unsigned _mx = ((s) >= (k)) ? (s) : (k); \
                          unsigned _mn = ((s) >= (k)) ? (k) : (s); \
                          (s) = _mx; (k) = _mn; }

__global__ void __launch_bounds__(64) zero_counts_kernel(float* __restrict__ cnt) {
    if (threadIdx.x < N_EXPERTS) cnt[threadIdx.x] = 0.0f;
}

__global__ void __launch_bounds__(THREADS) router_kernel(
    const float* __restrict__ logits,
    float* __restrict__ out_logits,
    float* __restrict__ out_w,
    float* __restrict__ out_id,
    float* __restrict__ out_cnt,
    int n_tokens)
{
    __shared__ __align__(16) float smem[TILE_TOKENS * ROW_DW];
    __shared__ unsigned hist[N_EXPERTS];

    const int tid = threadIdx.x;
    const int tile_base = blockIdx.x * TILE_TOKENS;
    int rows = n_tokens - tile_base;
    if (rows > TILE_TOKENS) rows = TILE_TOKENS;

    if (tid < N_EXPERTS) hist[tid] = 0u;

#if __has_builtin(__builtin_amdgcn_tensor_load_to_lds)
    // --- CDNA5 Tensor Data Mover: DMA the 256x64 fp32 tile into LDS with
    //     hardware row padding (4 dwords after every 64) for bank-conflict-free access.
    if (tid < 32) {  // one wave issues; TENSORcnt is per-wave
        const unsigned lds_base = (unsigned)(uintptr_t)(void*)smem;  // low 32 bits = LDS offset
        const unsigned long long ga =
            (unsigned long long)(uintptr_t)(logits + (size_t)tile_base * N_EXPERTS);
        const unsigned rem = (unsigned)(n_tokens - tile_base);  // rows remaining from tile start

        u32x4 g0;
        g0[0] = 1u;                                             // count=1, user descriptor
        g0[1] = lds_base;                                       // lds_addr [63:32]
        g0[2] = (unsigned)(ga & 0xFFFFFFFFull);                 // global_addr[31:0]
        g0[3] = (unsigned)((ga >> 32) & 0x1FFFFFFull)           // global_addr[56:32]
                | (2u << 30);                                   // type=2 ("image")

        i32x8 g1;
        g1[0] = (int)((2u << 16)      // data_size = 4B
                    | (1u << 20)      // pad_enable
                    | (5u << 22)      // pad_interval: 64 dwords
                    | (3u << 25));    // pad_amount: 4 dwords
        g1[1] = (int)((unsigned)N_EXPERTS << 16);               // tensor_dim0[15:0] @ bits 63:48
        g1[2] = (int)((rem & 0xFFFFu) << 16);                   // tensor_dim1[15:0] @ bits 95:80
        g1[3] = (int)(((rem >> 16) & 0xFFFFu)                   // tensor_dim1[31:16]
                    | ((unsigned)N_EXPERTS << 16));             // tile_dim0 = 64
        g1[4] = rows;                                           // tile_dim1; tile_dim2 = 0
        g1[5] = N_EXPERTS;                                      // tensor_dim0_stride = 64
        g1[6] = 0;
        g1[7] = 0;

        i32x4 g2 = {0, 0, 0, 0};
        i32x4 g3 = {0, 0, 0, 0};
#if defined(__clang_major__) && (__clang_major__ >= 23)
        i32x8 g4 = {0, 0, 0, 0, 0, 0, 0, 0};
        __builtin_amdgcn_tensor_load_to_lds(g0, g1, g2, g3, g4, 0);
#else
        __builtin_amdgcn_tensor_load_to_lds(g0, g1, g2, g3, 0);
#endif
#if __has_builtin(__builtin_amdgcn_s_wait_tensorcnt)
        __builtin_amdgcn_s_wait_tensorcnt(0);
#else
        asm volatile("s_wait_tensorcnt 0x0" ::: "memory");
#endif
    }
#else
    // --- Fallback staging: coalesced b128 global loads -> padded LDS stores
    {
        const size_t gbase = (size_t)tile_base * N_EXPERTS;
        const int tile_elems = rows * N_EXPERTS;
#pragma unroll
        for (int i = 0; i < 16; ++i) {
            int e = i * (THREADS * 4) + tid * 4;
            if (e < tile_elems) {
                f32x4 v = *(const f32x4*)(logits + gbase + e);
                int r = e >> 6, c = e & 63;
                *(f32x4*)(smem + r * ROW_DW + c) = v;
            }
        }
    }
#endif
    __syncthreads();

    // --- One token per thread: top-8 of 64 via packed-key insertion network.
    // Monotone uint transform of float bits; low 6 bits carry (63 - expert_idx)
    // so max-order == value-desc with ties broken by lower index (matches lax.top_k).
    if (tid < rows) {
        const int token = tile_base + tid;
        unsigned s0 = 0, s1 = 0, s2 = 0, s3 = 0, s4 = 0, s5 = 0, s6 = 0, s7 = 0;
        const f32x4* rowp = (const f32x4*)(smem + tid * ROW_DW);
#pragma unroll
        for (int i = 0; i < 16; ++i) {
            f32x4 v = rowp[i];  // ds_load_b128, conflict-free via 68-dword stride
#pragma unroll
            for (int c = 0; c < 4; ++c) {
                unsigned u = __float_as_uint(v[c]);
                u ^= ((unsigned)((int)u >> 31)) | 0x80000000u;   // order-preserving flip
                unsigned k = (u & 0xFFFFFFC0u) | (unsigned)(63 - (i * 4 + c));
                INS_STAGE(s0, k); INS_STAGE(s1, k); INS_STAGE(s2, k); INS_STAGE(s3, k);
                INS_STAGE(s4, k); INS_STAGE(s5, k); INS_STAGE(s6, k); INS_STAGE(s7, k);
            }
        }
        const unsigned keys[TOP_K] = {s0, s1, s2, s3, s4, s5, s6, s7};
        int   ids [TOP_K];
        float vals[TOP_K];
#pragma unroll
        for (int k = 0; k < TOP_K; ++k) {
            ids[k]  = 63 - (int)(keys[k] & 63u);
            vals[k] = smem[tid * ROW_DW + ids[k]];   // recover exact logits
        }
        // softmax denominator cancels under NORM_TOPK_PROB:
        // w_j = exp(l_j - m) / sum_top8 exp(l_i - m)
        const float m = vals[0];
        float ex[TOP_K];
        float sum = 0.0f;
#pragma unroll
        for (int k = 0; k < TOP_K; ++k) { ex[k] = __expf(vals[k] - m); sum += ex[k]; }
        const float inv = 1.0f / sum;

        f32x4 w_lo = { ex[0] * inv, ex[1] * inv, ex[2] * inv, ex[3] * inv };
        f32x4 w_hi = { ex[4] * inv, ex[5] * inv, ex[6] * inv, ex[7] * inv };
        f32x4 i_lo = { (float)ids[0], (float)ids[1], (float)ids[2], (float)ids[3] };
        f32x4 i_hi = { (float)ids[4], (float)ids[5], (float)ids[6], (float)ids[7] };
        const size_t ob = (size_t)token * TOP_K;
        __builtin_nontemporal_store(w_lo, (f32x4*)(out_w + ob));
        __builtin_nontemporal_store(w_hi, (f32x4*)(out_w + ob + 4));
        __builtin_nontemporal_store(i_lo, (f32x4*)(out_id + ob));
        __builtin_nontemporal_store(i_hi, (f32x4*)(out_id + ob + 4));

#pragma unroll
        for (int k = 0; k < TOP_K; ++k) atomicAdd(&hist[ids[k]], 1u);  // ds_add_u32
    }

    // --- Logits passthrough: LDS tile -> out_logits, coalesced NT b128 stores.
    {
        const size_t gbase = (size_t)tile_base * N_EXPERTS;
        const int tile_elems = rows * N_EXPERTS;
#pragma unroll
        for (int i = 0; i < 16; ++i) {
            int e = i * (THREADS * 4) + tid * 4;
            if (e < tile_elems) {
                int r = e >> 6, c = e & 63;
                f32x4 v = *(const f32x4*)(smem + r * ROW_DW + c);
                __builtin_nontemporal_store(v, (f32x4*)(out_logits + gbase + e));
            }
        }
    }

    __syncthreads();
    // --- Flush per-block histogram: one float atomic per expert per block.
    if (tid < N_EXPERTS) {
        unsigned h = hist[tid];
        if (h) atomicAdd(&out_cnt[tid], (float)h);   // global_atomic_add_f32
    }
}

extern "C" void kernel_launch(void* const* d_in, const int* in_sizes, int n_in,
                              void* d_out, int out_size, void* d_ws, size_t ws_size,
                              hipStream_t stream) {
    (void)n_in; (void)out_size; (void)d_ws; (void)ws_size;
    const float* logits = (const float*)d_in[0];
    const int n_tokens = in_sizes[0] / N_EXPERTS;

    float* out_logits = (float*)d_out;
    float* out_w      = out_logits + (size_t)n_tokens * N_EXPERTS;
    float* out_id     = out_w      + (size_t)n_tokens * TOP_K;
    float* out_cnt    = out_id     + (size_t)n_tokens * TOP_K;

    zero_counts_kernel<<<1, 64, 0, stream>>>(out_cnt);  // deterministic across graph replays

    const int blocks = (n_tokens + TILE_TOKENS - 1) / TILE_TOKENS;
    router_kernel<<<blocks, THREADS, 0, stream>>>(logits, out_logits, out_w, out_id,
                                                  out_cnt, n_tokens);
}